// OpticalFlowWarpLoss_87282325389782
// MI455X (gfx1250) — compile-verified
//
#include <hip/hip_runtime.h>
#include <hip/hip_bf16.h>

// Problem constants (fixed by the reference).
constexpr int kB = 16, kC = 3, kH = 512, kW = 512;
constexpr int kHW = kH * kW;               // 262144 = 2^18
constexpr int kNPix = kB * kHW;            // 4,194,304
constexpr int kThreads = 256;              // 8 wave32s per block
constexpr int kBlocks = kNPix / kThreads;  // 16384
constexpr float kInvCount = 1.0f / (float)(kB * kC * kHW); // 1/12,582,912

typedef __attribute__((ext_vector_type(2))) float v2f;
typedef __attribute__((ext_vector_type(8))) float v8f;

// ---------------------------------------------------------------------------
// Kernel 1: bilinear warp + |warped - x_hat|, per-block partial sums.
// One thread per pixel (b,h,w); 3 channels per thread.
// flow / x_hat are single-use streams -> non-temporal loads (keep x in L2,
// it is re-read ~4x by the gather). x gathers use default RT policy.
// ---------------------------------------------------------------------------
__global__ __launch_bounds__(kThreads)
void warp_l1_partial(const float* __restrict__ flow,
                     const float* __restrict__ x,
                     const float* __restrict__ xhat,
                     float* __restrict__ partial) {
    const int t = blockIdx.x * kThreads + threadIdx.x;   // < kNPix by construction
    const int b = t >> 18;          // / (H*W)
    const int r = t & (kHW - 1);    // h*W + w
    const int h = r >> 9;           // / W
    const int w = r & (kW - 1);

    // flow[:,0] shifts the ROW index but feeds the WIDTH coordinate (and vice
    // versa) — preserve the reference's channel assignment exactly.
    const float f0 = __builtin_nontemporal_load(&flow[(b * 2 + 0) * kHW + r]);
    const float f1 = __builtin_nontemporal_load(&flow[(b * 2 + 1) * kHW + r]);

    const float gx = 2.0f * ((float)h + f0) / (float)(kH - 1) - 1.0f;
    const float gy = 2.0f * ((float)w + f1) / (float)(kW - 1) - 1.0f;
    const float wp = ((gx + 1.0f) * (float)kW - 1.0f) * 0.5f;  // width coord
    const float hp = ((gy + 1.0f) * (float)kH - 1.0f) * 0.5f;  // height coord

    const float w0f = floorf(wp);
    const float h0f = floorf(hp);
    const float dw = wp - w0f;
    const float dh = hp - h0f;
    const int w0i = (int)w0f, h0i = (int)h0f;
    const int w1i = w0i + 1, h1i = h0i + 1;

    const bool vh0 = (h0i >= 0) & (h0i < kH);
    const bool vh1 = (h1i >= 0) & (h1i < kH);
    const bool vw0 = (w0i >= 0) & (w0i < kW);
    const bool vw1 = (w1i >= 0) & (w1i < kW);

    const int hc0 = min(max(h0i, 0), kH - 1);
    const int hc1 = min(max(h1i, 0), kH - 1);
    const int wc0 = min(max(w0i, 0), kW - 1);
    const int wc1 = min(max(w1i, 0), kW - 1);

    const float wt00 = (1.0f - dh) * (1.0f - dw);
    const float wt01 = (1.0f - dh) * dw;
    const float wt10 = dh * (1.0f - dw);
    const float wt11 = dh * dw;

    float lsum = 0.0f;
    const float* xb = x + (size_t)b * kC * kHW;
#pragma unroll
    for (int c = 0; c < kC; ++c) {
        const float* xp = xb + c * kHW;
        const float v00 = (vh0 & vw0) ? xp[hc0 * kW + wc0] : 0.0f;
        const float v01 = (vh0 & vw1) ? xp[hc0 * kW + wc1] : 0.0f;
        const float v10 = (vh1 & vw0) ? xp[hc1 * kW + wc0] : 0.0f;
        const float v11 = (vh1 & vw1) ? xp[hc1 * kW + wc1] : 0.0f;
        const float warped = v00 * wt00 + v01 * wt01 + v10 * wt10 + v11 * wt11;
        const float xh = __builtin_nontemporal_load(&xhat[((size_t)b * kC + c) * kHW + r]);
        lsum += fabsf(warped - xh);
    }

    // wave32 reduction, then cross-wave via LDS (fixed order -> deterministic)
#pragma unroll
    for (int off = 16; off > 0; off >>= 1)
        lsum += __shfl_down(lsum, off, 32);

    __shared__ float sdata[kThreads / 32];
    const int lane = threadIdx.x & 31;
    const int wid = threadIdx.x >> 5;
    if (lane == 0) sdata[wid] = lsum;
    __syncthreads();
    if (threadIdx.x == 0) {
        float s = 0.0f;
#pragma unroll
        for (int i = 0; i < kThreads / 32; ++i) s += sdata[i];
        partial[blockIdx.x] = s;
    }
}

// ---------------------------------------------------------------------------
// Kernel 2: deterministic reduction of kBlocks partials via WMMA.
// D = A x B + C with B = ones(4x16): every element of A is added into every
// column of D, so the A-layout mapping is irrelevant. Accumulates 64 values
// per v_wmma_f32_16x16x4_f32 issue, exact f32 RNE.
// Single wave of 32 (EXEC all ones before/through the WMMA loop).
// ---------------------------------------------------------------------------
__global__ __launch_bounds__(32)
void reduce_partials_wmma(const float* __restrict__ partial,
                          float* __restrict__ out,
                          int n, float inv_count) {
    const int lane = threadIdx.x;
    v8f acc = {};
    v2f ones;
    ones.x = 1.0f;
    ones.y = 1.0f;
    for (int i = 0; i < n; i += 64) {
        v2f a;
        a.x = partial[i + lane];
        a.y = partial[i + lane + 32];
        acc = __builtin_amdgcn_wmma_f32_16x16x4_f32(
            /*neg_a=*/false, a, /*neg_b=*/false, ones,
            /*c_mod=*/(short)0, acc, /*reuse_a=*/false, /*reuse_b=*/false);
    }
    // Column 0 of D lives in lane 0 (M=0..7 in acc[0..7]) and lane 16
    // (M=8..15). Their sum over the 8 accumulator VGPRs is the grand total.
    float s = acc[0] + acc[1] + acc[2] + acc[3] +
              acc[4] + acc[5] + acc[6] + acc[7];
    s += __shfl_down(s, 16, 32);
    if (lane == 0) out[0] = s * inv_count;
}

extern "C" void kernel_launch(void* const* d_in, const int* in_sizes, int n_in,
                              void* d_out, int out_size, void* d_ws, size_t ws_size,
                              hipStream_t stream) {
    const float* flow = (const float*)d_in[0];
    const float* x    = (const float*)d_in[1];
    const float* xhat = (const float*)d_in[2];
    float* out = (float*)d_out;
    float* partial = (float*)d_ws;  // kBlocks floats = 64 KB

    warp_l1_partial<<<kBlocks, kThreads, 0, stream>>>(flow, x, xhat, partial);
    reduce_partials_wmma<<<1, 32, 0, stream>>>(partial, out, kBlocks, kInvCount);
}